// DurationCalculator_1271310319651
// MI455X (gfx1250) — compile-verified
//
#include <hip/hip_runtime.h>
#include <float.h>

// ---------------------------------------------------------------------------
// DurationCalculator for MI455X (gfx1250, wave32).
// Pure bandwidth problem (272 MB traffic, ~12us at 23.3 TB/s). No matrix ops
// exist in this op, so the CDNA5 paths used are: async global->LDS streaming
// (ASYNCcnt), wave32 shuffle reductions, global prefetch.
// ---------------------------------------------------------------------------

#define AS1 __attribute__((address_space(1)))
#define AS3 __attribute__((address_space(3)))

typedef int v4i __attribute__((vector_size(16)));  // 4 x i32, b128 payload

#if defined(__HIP_DEVICE_COMPILE__) && defined(__gfx1250__) && \
    __has_builtin(__builtin_amdgcn_global_load_async_to_lds_b128) && \
    __has_builtin(__builtin_amdgcn_s_wait_asynccnt)
#define USE_ASYNC_LDS 1
#else
#define USE_ASYNC_LDS 0
#endif

namespace {
constexpr int B = 16, H = 16, L = 1024, T = 256;
constexpr int SPLIT = 4;          // blocks per (b,h) along L
constexpr int ROWS  = L / SPLIT;  // 256 rows handled per block in pass 1
}

// ---------------------------------------------------------------------------
// Kernel 0: zero the output histogram and the per-head score accumulators.
// ---------------------------------------------------------------------------
__global__ void k_init(float* __restrict__ out, float* __restrict__ scores) {
  const int i = blockIdx.x * blockDim.x + threadIdx.x;
  if (i < B * T) out[i] = 0.0f;
  if (i < H)     scores[i] = 0.0f;
}

// ---------------------------------------------------------------------------
// Kernel 1: scores[h] += sum over (b, l-chunk) of max_t att[b,h,l,t].
// One block per (b,h,chunk); 8 waves; each wave reduces one row at a time
// (2 x float4 per lane, butterfly max across 32 lanes).
// Data is streamed through LDS with double-buffered async b128 copies.
// ---------------------------------------------------------------------------
__global__ void k_scores(const float* __restrict__ att, float* __restrict__ scores) {
  const int bid  = blockIdx.x;
  const int s    = bid % SPLIT;
  const int bh   = bid / SPLIT;
  const int h    = bh % H;
  const float* base = att + (size_t)bh * (L * T) + (size_t)s * ROWS * T;
  const int tid  = threadIdx.x;         // 0..255
  const int wave = tid >> 5;
  const int lane = tid & 31;
  float acc = 0.0f;

#if USE_ASYNC_LDS
  __shared__ __align__(16) float lbuf[2][16 * T];   // 2 x 16 KB tiles (16 rows)
  constexpr int NT = ROWS / 16;                     // 16 tiles per block

  auto issue = [&](int tile, int buf) {
    const float* gt = base + (size_t)tile * (16 * T);
    // j-major layout: per instruction, lanes cover 512B contiguous.
#pragma unroll
    for (int j = 0; j < 4; ++j) {
      const int ofs = tid * 4 + j * 1024;           // floats
      __builtin_amdgcn_global_load_async_to_lds_b128(
          (AS1 v4i*)(gt + ofs), (AS3 v4i*)(&lbuf[buf][ofs]),
          /*offset=*/0, /*cpol=*/0);
    }
  };

  issue(0, 0);
  for (int tile = 0; tile < NT; ++tile) {
    if (tile + 1 < NT) {
      issue(tile + 1, (tile + 1) & 1);
      __builtin_amdgcn_s_wait_asynccnt(4);  // oldest 4 (this tile) complete
    } else {
      __builtin_amdgcn_s_wait_asynccnt(0);
    }
    __syncthreads();
    const float* tb = &lbuf[tile & 1][0];
#pragma unroll
    for (int rr = 0; rr < 2; ++rr) {
      const int r = wave + rr * 8;                  // 16 rows / 8 waves
      const float4* rp = (const float4*)(tb + r * T);
      float4 a = rp[lane];
      float4 c = rp[lane + 32];
      float m = fmaxf(fmaxf(fmaxf(a.x, a.y), fmaxf(a.z, a.w)),
                      fmaxf(fmaxf(c.x, c.y), fmaxf(c.z, c.w)));
#pragma unroll
      for (int sh = 16; sh >= 1; sh >>= 1) m = fmaxf(m, __shfl_xor(m, sh, 32));
      acc += m;                                     // identical across lanes
    }
    __syncthreads();
  }
#else
  for (int l = wave; l < ROWS; l += 8) {
    const float4* rp = (const float4*)(base + (size_t)l * T);
    float4 a = rp[lane];
    float4 c = rp[lane + 32];
    float m = fmaxf(fmaxf(fmaxf(a.x, a.y), fmaxf(a.z, a.w)),
                    fmaxf(fmaxf(c.x, c.y), fmaxf(c.z, c.w)));
#pragma unroll
    for (int sh = 16; sh >= 1; sh >>= 1) m = fmaxf(m, __shfl_xor(m, sh, 32));
    acc += m;
  }
#endif

  __shared__ float sAcc[8];
  if (lane == 0) sAcc[wave] = acc;
  __syncthreads();
  if (tid == 0) {
    float sum = 0.0f;
#pragma unroll
    for (int w = 0; w < 8; ++w) sum += sAcc[w];
    atomicAdd(&scores[h], sum);
  }
}

// ---------------------------------------------------------------------------
// Kernel 2: head = argmax_h scores[h] (first occurrence on ties). One wave.
// ---------------------------------------------------------------------------
__global__ void k_head(const float* __restrict__ scores, int* __restrict__ headp) {
  const int lane = threadIdx.x;  // 32 threads
  float v = (lane < H) ? scores[lane] : -FLT_MAX;
  int   i = (lane < H) ? lane : 0x7fffffff;
#pragma unroll
  for (int sh = 16; sh >= 1; sh >>= 1) {
    float ov = __shfl_xor(v, sh, 32);
    int   oi = __shfl_xor(i, sh, 32);
    if (ov > v || (ov == v && oi < i)) { v = ov; i = oi; }
  }
  if (lane == 0) *headp = i;
}

// ---------------------------------------------------------------------------
// Kernel 3: per-row masked argmax over t < ilen[b] for the selected head,
// then histogram: out[b, amax] += 1 for rows l < olen[b].
// One wave per row; tie-break = lowest index (matches jnp.argmax).
// ---------------------------------------------------------------------------
__global__ void k_dur(const float* __restrict__ att, const int* __restrict__ ilens,
                      const int* __restrict__ olens, const int* __restrict__ headp,
                      float* __restrict__ out) {
  const int head  = *headp;
  const int bid   = blockIdx.x;            // B * (L/64) blocks
  const int b     = bid >> 4;
  const int lbase = (bid & 15) * 64;
  const int wave  = threadIdx.x >> 5;
  const int lane  = threadIdx.x & 31;
  const int ilen  = ilens[b];
  const int olen  = olens[b];
  const float* bbase = att + (size_t)(b * H + head) * (L * T);

  for (int l = lbase + wave; l < lbase + 64; l += 8) {
    if (l >= olen) continue;               // zero weight beyond olen
    const float* row = bbase + (size_t)l * T;
    __builtin_prefetch(row + 8 * T, 0, 0); // stream hint: 8 rows ahead
    const float4* rp = (const float4*)row;
    float4 a = rp[lane];                   // t = 4*lane + j
    float4 c = rp[lane + 32];              // t = 128 + 4*lane + j
    const float va[8] = {a.x, a.y, a.z, a.w, c.x, c.y, c.z, c.w};
    float bv = -FLT_MAX;
    int   bi = 0;
    const int t0 = lane * 4;
#pragma unroll
    for (int j = 0; j < 8; ++j) {
      const int t = (j < 4) ? (t0 + j) : (128 + t0 + (j - 4));
      if (t < ilen && va[j] > bv) { bv = va[j]; bi = t; }  // strict > keeps first max
    }
#pragma unroll
    for (int sh = 16; sh >= 1; sh >>= 1) {
      float ov = __shfl_xor(bv, sh, 32);
      int   oi = __shfl_xor(bi, sh, 32);
      if (ov > bv || (ov == bv && oi < bi)) { bv = ov; bi = oi; }
    }
    if (lane == 0) atomicAdd(&out[b * T + bi], 1.0f);
  }
}

// ---------------------------------------------------------------------------
extern "C" void kernel_launch(void* const* d_in, const int* in_sizes, int n_in,
                              void* d_out, int out_size, void* d_ws, size_t ws_size,
                              hipStream_t stream) {
  (void)in_sizes; (void)n_in; (void)out_size; (void)ws_size;
  const float* att   = (const float*)d_in[0];
  const int*   ilens = (const int*)d_in[1];
  const int*   olens = (const int*)d_in[2];
  float*       out   = (float*)d_out;
  float*       scores = (float*)d_ws;                 // 16 floats
  int*         headp  = (int*)((char*)d_ws + 64);     // 1 int

  k_init  <<<16, 256, 0, stream>>>(out, scores);
  k_scores<<<B * H * SPLIT, 256, 0, stream>>>(att, scores);
  k_head  <<<1, 32, 0, stream>>>(scores, headp);
  k_dur   <<<B * (L / 64), 256, 0, stream>>>(att, ilens, olens, headp, out);
}